// SplineConv_19731079758626
// MI455X (gfx1250) — compile-verified
//
#include <hip/hip_runtime.h>

typedef __attribute__((ext_vector_type(2))) float v2f;
typedef __attribute__((ext_vector_type(8))) float v8f;

#define N_NODES 100000
#define N_EDGES 1250000
#define D 64
#define KS 2

// ---------------------------------------------------------------------------
// Fused GEMM: one wave computes one 16x16 tile of X(100000x64) @ B(64x64)
// for B in {W[0], W[1], W_root}. Uses V_WMMA_F32_16X16X4_F32 (fp32-exact).
// xw0/xw1 -> workspace, X@W_root -> d_out (finalize adds agg/cnt on top).
// ---------------------------------------------------------------------------
__global__ __launch_bounds__(256) void spline_gemm_wmma(
    const float* __restrict__ X,    // [N,64] row-major
    const float* __restrict__ W,    // [2,64,64]
    const float* __restrict__ Wr,   // [64,64]
    float* __restrict__ xw0,        // [N,64]
    float* __restrict__ xw1,        // [N,64]
    float* __restrict__ outRoot)    // [N,64] (= d_out)
{
    const int wave  = (blockIdx.x * blockDim.x + threadIdx.x) >> 5;
    const int lane  = threadIdx.x & 31;
    const int mtile = wave / 12;            // 0..6249  (6250*16 = 100000 rows)
    const int nt    = wave - mtile * 12;    // 0..11
    const int g     = nt >> 2;              // which B matrix
    const int ntile = nt & 3;               // 0..3 (4 * 16 = 64 cols)

    const float* Bmat = (g == 0) ? W : (g == 1) ? (W + D * D) : Wr;
    float*       Out  = (g == 0) ? xw0 : (g == 1) ? xw1 : outRoot;

    const int half  = lane >> 4;            // 0: lanes 0-15, 1: lanes 16-31
    const int l15   = lane & 15;
    const int row   = mtile * 16 + l15;     // A-matrix M row for this lane
    const int col   = ntile * 16 + l15;     // B/C/D N column for this lane
    const int khalf = half * 2;             // K sub-offset per ISA A/B layout

    const float* arow = X + (size_t)row * D + khalf;      // A[M=row][K=k+khalf, +1]
    const float* bcol = Bmat + (size_t)khalf * D + col;   // B[K=k+khalf][N=col]

    v8f c = {};
#pragma unroll
    for (int k = 0; k < D; k += 4) {
        v2f a = *reinterpret_cast<const v2f*>(arow + k);  // contiguous K pair
        v2f b;
        b.x = bcol[(size_t)k * D];                        // B[k+khalf  ][col]
        b.y = bcol[(size_t)k * D + D];                    // B[k+khalf+1][col]
        // 8 args: (neg_a, A, neg_b, B, c_mod, C, reuse_a, reuse_b)
        c = __builtin_amdgcn_wmma_f32_16x16x4_f32(false, a, false, b,
                                                  (short)0, c, false, false);
    }

    // D layout: VGPR i -> M = i + 8*half, N = l15
    float* op = Out + (size_t)(mtile * 16 + half * 8) * D + col;
#pragma unroll
    for (int i = 0; i < 8; ++i)
        op[(size_t)i * D] = c[i];
}

// ---------------------------------------------------------------------------
// Zero agg + cnt (contiguous in workspace).
// ---------------------------------------------------------------------------
__global__ void spline_zero(float* __restrict__ p, int n) {
    int i = blockIdx.x * blockDim.x + threadIdx.x;
    if (i < n) p[i] = 0.0f;
}

// ---------------------------------------------------------------------------
// Edge phase: 32 lanes per edge, 2 channels (float2) per lane.
// xw tables (51 MB) fit in the 192 MB L2, so gathers stay on-chip.
// ---------------------------------------------------------------------------
__global__ __launch_bounds__(256) void spline_edges(
    const int*   __restrict__ ei,   // [2,E] (src row 0, dst row 1)
    const float* __restrict__ ef,   // [E]
    const float* __restrict__ xw0,
    const float* __restrict__ xw1,
    float* __restrict__ agg,        // [N,64]
    float* __restrict__ cnt)        // [N]
{
    long long gt = (long long)blockIdx.x * blockDim.x + threadIdx.x;
    int e    = (int)(gt >> 5);
    int lane = (int)(gt & 31);
    if (e >= N_EDGES) return;

    int   src = ei[e];
    int   dst = ei[N_EDGES + e];
    float u   = ef[e];

    // Linear B-spline basis: v = u*(K-1); two active knots i0, i1.
    float v    = u * (float)(KS - 1);
    float bf   = floorf(v);
    float frac = v - bf;
    int i0 = (int)bf;
    i0 = i0 < 0 ? 0 : (i0 > KS - 1 ? KS - 1 : i0);   // clamp for memory safety
    int i1 = i0 + 1; if (i1 > KS - 1) i1 = KS - 1;

    const float* t0 = (i0 == 0) ? xw0 : xw1;
    const float* t1 = (i1 == 0) ? xw0 : xw1;

    int c0 = lane * 2;
    float2 p0 = *reinterpret_cast<const float2*>(t0 + (size_t)src * D + c0);
    float2 p1 = *reinterpret_cast<const float2*>(t1 + (size_t)src * D + c0);

    float w0 = 1.0f - frac, w1 = frac;
    float mx = w0 * p0.x + w1 * p1.x;
    float my = w0 * p0.y + w1 * p1.y;

    float* ap = agg + (size_t)dst * D + c0;
    atomicAdd(ap,     mx);
    atomicAdd(ap + 1, my);
    if (lane == 0) atomicAdd(cnt + dst, 1.0f);
}

// ---------------------------------------------------------------------------
// out = agg / max(cnt,1) + root   (root already resident in d_out)
// ---------------------------------------------------------------------------
__global__ void spline_finalize(const float* __restrict__ agg,
                                const float* __restrict__ cnt,
                                float* __restrict__ out) {
    int i = blockIdx.x * blockDim.x + threadIdx.x;
    if (i < N_NODES * D) {
        int n = i >> 6;
        out[i] = agg[i] / fmaxf(cnt[n], 1.0f) + out[i];
    }
}

extern "C" void kernel_launch(void* const* d_in, const int* in_sizes, int n_in,
                              void* d_out, int out_size, void* d_ws, size_t ws_size,
                              hipStream_t stream) {
    const float* X  = (const float*)d_in[0];   // node_feature [100000,64] f32
    const int*   ei = (const int*)d_in[1];     // edge_index   [2,1250000] int
    const float* ef = (const float*)d_in[2];   // edge_feature [1250000,1] f32
    const float* W  = (const float*)d_in[3];   // W      [2,64,64] f32
    const float* Wr = (const float*)d_in[4];   // W_root [64,64]   f32
    float* out = (float*)d_out;

    float* ws  = (float*)d_ws;
    const size_t ND = (size_t)N_NODES * D;
    float* xw0 = ws;             // [N,64]
    float* xw1 = ws + ND;        // [N,64]
    float* agg = ws + 2 * ND;    // [N,64]
    float* cnt = ws + 3 * ND;    // [N]      (contiguous after agg)

    // 1) zero agg + cnt (they are contiguous)
    int nz = N_NODES * D + N_NODES;
    spline_zero<<<(nz + 255) / 256, 256, 0, stream>>>(agg, nz);

    // 2) fused WMMA GEMMs: xw0, xw1 -> ws ; X@W_root -> d_out
    //    waves = 6250 mtiles * 12 ntiles = 75000 -> 9375 blocks of 8 waves
    spline_gemm_wmma<<<9375, 256, 0, stream>>>(X, W, Wr, xw0, xw1, out);

    // 3) edge gather/blend/scatter: 32 threads per edge
    long long tot = (long long)N_EDGES * 32;
    spline_edges<<<(int)((tot + 255) / 256), 256, 0, stream>>>(ei, ef, xw0, xw1, agg, cnt);

    // 4) out = agg/max(cnt,1) + root
    spline_finalize<<<(N_NODES * D + 255) / 256, 256, 0, stream>>>(agg, cnt, out);
}